// QLoss_10960756539998
// MI455X (gfx1250) — compile-verified
//
#include <hip/hip_runtime.h>

// ---------------------------------------------------------------------------
// QLoss: q = (in < 1e-3) ? (1-in)*100 : (in > tg ? |in|/|tg| : |tg|/|in|)
// out  = mean(q) over N = 2^25 f32 elements.
//
// Roofline: 2 * 2^25 * 4 B = 256 MiB streamed once @ 23.3 TB/s ~ 11.5 us.
// Pure HBM-bound: ext-vector float4 (global_load_b128) non-temporal loads,
// 2x unrolled grid-stride loop (4 b128 loads in flight per lane), trivial
// VALU math with v_rcp_f32 + 1 Newton step. Wave reduction uses
// V_WMMA_F32_16X16X4_F32 with B = ones as a one-instruction 64->16
// cross-lane adder on the XDL pipe. Deterministic: fixed-order sums,
// no float atomics (graph-replay stable).
// ---------------------------------------------------------------------------

typedef __attribute__((ext_vector_type(2))) float v2f;
typedef __attribute__((ext_vector_type(4))) float v4f;   // clang vector: OK for NT builtin
typedef __attribute__((ext_vector_type(8))) float v8f;

#define QL_BLOCK 256
#define QL_GRID  2048

__device__ __forceinline__ float qerr(float in, float tg) {
    float ai = __builtin_fabsf(in);
    float at = __builtin_fabsf(tg);
    bool  gt = in > tg;
    float num = gt ? ai : at;
    float den = gt ? at : ai;
    // fast reciprocal + one Newton-Raphson step: r1 = r*(2 - den*r)
    float r = __builtin_amdgcn_rcpf(den);
    r = r * __builtin_fmaf(-den, r, 2.0f);
    float ratio = num * r;
    return (in < 0.001f) ? (1.0f - in) * 100.0f : ratio;
}

__device__ __forceinline__ float qerr4(v4f a, v4f b) {
    return (qerr(a.x, b.x) + qerr(a.z, b.z)) + (qerr(a.y, b.y) + qerr(a.w, b.w));
}

// Wave32 sum of 2 partials per lane (64 values) using one WMMA:
//   A(16x4) = per-lane partials  [lane<16: K=0,1 | lane>=16: K=2,3]
//   B(4x16) = ones  ->  D[m][n] = rowsum(m)  (lane-invariant along n)
// Sum D's 8 VGPRs per lane, then total = t@lane0 (rows 0-7) + t@lane16 (rows 8-15).
__device__ __forceinline__ float wave_reduce_wmma(float s0, float s1) {
    v2f a; a[0] = s0;   a[1] = s1;
    v2f b; b[0] = 1.0f; b[1] = 1.0f;
    v8f c = {};
    c = __builtin_amdgcn_wmma_f32_16x16x4_f32(
            /*neg_a=*/false, a, /*neg_b=*/false, b,
            /*c_mod=*/(short)0, c, /*reuse_a=*/false, /*reuse_b=*/false);
    float t = ((c[0] + c[1]) + (c[2] + c[3])) + ((c[4] + c[5]) + (c[6] + c[7]));
    return __shfl(t, 0, 32) + __shfl(t, 16, 32);
}

__device__ __forceinline__ void block_reduce_store(float wave_total,
                                                   float* __restrict__ dst,
                                                   float scale) {
    __shared__ float lds[QL_BLOCK / 32];
    const int lane = threadIdx.x & 31;
    const int wid  = threadIdx.x >> 5;
    if (lane == 0) lds[wid] = wave_total;
    __syncthreads();
    if (threadIdx.x == 0) {
        float t = 0.0f;
        const int nw = blockDim.x >> 5;
        for (int i = 0; i < nw; ++i) t += lds[i];   // fixed order: deterministic
        *dst = t * scale;
    }
}

// Pass 1: grid-stride over float4s, per-block partial -> d_ws[blockIdx.x]
__global__ void qloss_partial(const float* __restrict__ in,
                              const float* __restrict__ tg,
                              float* __restrict__ partial,
                              long long n4, long long n) {
    const long long idx    = (long long)blockIdx.x * blockDim.x + threadIdx.x;
    const long long stride = (long long)gridDim.x * blockDim.x;
    const v4f* __restrict__ in4 = (const v4f*)in;
    const v4f* __restrict__ tg4 = (const v4f*)tg;

    float s0 = 0.0f, s1 = 0.0f, s2 = 0.0f, s3 = 0.0f;

    // 2x unrolled grid-stride: 4 non-temporal b128 loads in flight per iter.
    long long i = idx;
    for (; i + stride < n4; i += 2 * stride) {
        v4f a0 = __builtin_nontemporal_load(&in4[i]);
        v4f b0 = __builtin_nontemporal_load(&tg4[i]);
        v4f a1 = __builtin_nontemporal_load(&in4[i + stride]);
        v4f b1 = __builtin_nontemporal_load(&tg4[i + stride]);
        s0 += qerr4(a0, b0);
        s1 += qerr4(a1, b1);
    }
    if (i < n4) {
        v4f a0 = __builtin_nontemporal_load(&in4[i]);
        v4f b0 = __builtin_nontemporal_load(&tg4[i]);
        s2 += qerr4(a0, b0);
    }
    // Scalar epilogue for N % 4 != 0 (dead for N = 2^25): first (n - n4*4)
    // global threads take one element each. No loop -> no SALU-float sled.
    {
        long long e = n4 * 4 + idx;
        if (e < n) s3 += qerr(in[e], tg[e]);
    }

    // control flow reconverged here: EXEC all-ones for WMMA
    float wtot = wave_reduce_wmma(s0 + s2, s1 + s3);
    block_reduce_store(wtot, &partial[blockIdx.x], 1.0f);
}

// Pass 2: single block folds the per-block partials and applies 1/N.
__global__ void qloss_final(const float* __restrict__ partial, int np,
                            float* __restrict__ out, float inv_n) {
    float s0 = 0.0f, s1 = 0.0f;
    for (int i = threadIdx.x; i < np; i += blockDim.x) {
        if (i & 1) s1 += partial[i]; else s0 += partial[i];
    }
    float wtot = wave_reduce_wmma(s0, s1);
    block_reduce_store(wtot, out, inv_n);
}

extern "C" void kernel_launch(void* const* d_in, const int* in_sizes, int n_in,
                              void* d_out, int out_size, void* d_ws, size_t ws_size,
                              hipStream_t stream) {
    const float* in = (const float*)d_in[0];
    const float* tg = (const float*)d_in[1];
    float* out = (float*)d_out;
    float* ws  = (float*)d_ws;

    const long long n  = (long long)in_sizes[0];
    const long long n4 = n >> 2;

    int nb = QL_GRID;
    if ((size_t)nb * sizeof(float) > ws_size)            // never overrun scratch
        nb = (int)(ws_size / sizeof(float));
    if (nb < 1) nb = 1;

    qloss_partial<<<nb, QL_BLOCK, 0, stream>>>(in, tg, ws, n4, n);
    qloss_final<<<1, QL_BLOCK, 0, stream>>>(ws, nb, out, 1.0f / (float)n);
}